// lstm_32779190403342
// MI455X (gfx1250) — compile-verified
//
#include <hip/hip_runtime.h>
#include <hip/hip_bf16.h>

#define T_IN 40
#define EMB  20
#define HID  20
#define G4   80   // 4*HID

typedef __attribute__((ext_vector_type(2))) float v2f;
typedef __attribute__((ext_vector_type(8))) float v8f;

__device__ __forceinline__ float sig_(float x) { return 1.f / (1.f + __expf(-x)); }
__device__ __forceinline__ float th_(float x)  { float e = __expf(-2.f * x); return (1.f - e) / (1.f + e); }

// ---------------------------------------------------------------------------
// Kernel A: fuse the two input projections into one weight matrix.
// W_fused[80,40] = W_ih0[80,20] @ W_inp[20,40]
// b_fused[80]    = W_ih0 @ b_inp + b_ih0 + b_hh0
// ---------------------------------------------------------------------------
__global__ void fuse_weights(const float* __restrict__ W_inp,
                             const float* __restrict__ b_inp,
                             const float* __restrict__ W_ih0,
                             const float* __restrict__ b_ih0,
                             const float* __restrict__ b_hh0,
                             float* __restrict__ Wf,
                             float* __restrict__ bf) {
    int tid = threadIdx.x;
    for (int idx = tid; idx < G4 * T_IN; idx += blockDim.x) {
        int g = idx / T_IN, j = idx % T_IN;
        float s = 0.f;
        #pragma unroll
        for (int e = 0; e < EMB; ++e)
            s = fmaf(W_ih0[g * EMB + e], W_inp[e * T_IN + j], s);
        Wf[idx] = s;
    }
    if (tid < G4) {
        float s = b_ih0[tid] + b_hh0[tid];
        #pragma unroll
        for (int e = 0; e < EMB; ++e)
            s = fmaf(W_ih0[tid * EMB + e], b_inp[e], s);
        bf[tid] = s;
    }
}

// ---------------------------------------------------------------------------
// Kernel B: pre0[T,80] = in_states[T,40] @ Wf.T + bf   via V_WMMA_F32_16X16X4_F32.
// One wave per 16(M)x16(N) output tile, K-loop of 10 WMMAs (K=40).
// A 16x4 f32 layout: lane L holds M=L%16, K pair at (L>>4)*2 (2 VGPRs).
// B 4x16 f32 layout: symmetric with N in lane%16.
// C/D: VGPR r -> row m0 + r + 8*(L>>4), col n0 + L%16.
// ---------------------------------------------------------------------------
__global__ __launch_bounds__(256)
void pre0_gemm(const float* __restrict__ X,   // [T,40]
               const float* __restrict__ Wf,  // [80,40]
               const float* __restrict__ bf,  // [80]
               float* __restrict__ pre0,      // [T,80]
               int T, int numTiles) {
    int wave = threadIdx.x >> 5;
    int lane = threadIdx.x & 31;
    int tile = blockIdx.x * (blockDim.x >> 5) + wave;
    if (tile >= numTiles) return;

    int mt = tile / 5, nt = tile % 5;
    int m0 = mt * 16, n0 = nt * 16;
    int lmod = lane & 15, lhi = lane >> 4;

    int mrow = m0 + lmod; if (mrow >= T) mrow = T - 1;   // clamp: only feeds unstored rows
    int gcol = n0 + lmod;

    const float* arow = X  + (size_t)mrow * T_IN + lhi * 2;
    const float* brow = Wf + (size_t)gcol * T_IN + lhi * 2;

    v8f acc = {};
#if __has_builtin(__builtin_amdgcn_wmma_f32_16x16x4_f32)
    #pragma unroll
    for (int k0 = 0; k0 < T_IN; k0 += 4) {
        v2f a, b;
        a.x = arow[k0]; a.y = arow[k0 + 1];
        b.x = brow[k0]; b.y = brow[k0 + 1];
        acc = __builtin_amdgcn_wmma_f32_16x16x4_f32(
            /*neg_a=*/false, a, /*neg_b=*/false, b,
            /*c_mod=*/(short)0, acc, /*reuse_a=*/false, /*reuse_b=*/false);
    }
#else
    #pragma unroll
    for (int r = 0; r < 8; ++r) {
        int row = m0 + r + lhi * 8; if (row >= T) row = T - 1;
        const float* xr = X + (size_t)row * T_IN;
        float s = 0.f;
        for (int k = 0; k < T_IN; ++k) s = fmaf(xr[k], Wf[(size_t)gcol * T_IN + k], s);
        acc[r] = s;
    }
#endif
    float bias = bf[gcol];
    #pragma unroll
    for (int r = 0; r < 8; ++r) {
        int row = m0 + r + lhi * 8;
        if (row < T) pre0[(size_t)row * G4 + gcol] = acc[r] + bias;
    }
}

// ---------------------------------------------------------------------------
// Kernel C: sequential LSTM scan, single workgroup, layers software-pipelined.
// Iteration n: waves 0-2 (tid<80) compute layer0 gates for step n while
// waves 3-5 (tid 96..175) compute layer1 gates for step n-1. Two barriers/step.
// pre0 row is prefetched one step ahead. out[t] reduced by shfl_xor in wave 1.
// ---------------------------------------------------------------------------
__global__ __launch_bounds__(192, 1)
void lstm_scan(const float* __restrict__ pre0,   // [T,80]
               const float* __restrict__ W_hh0,  // [80,20]
               const float* __restrict__ W_ih1,  // [80,20]
               const float* __restrict__ W_hh1,  // [80,20]
               const float* __restrict__ b_ih1,
               const float* __restrict__ b_hh1,
               const float* __restrict__ W_out,  // [20]
               const float* __restrict__ b_out,  // [1]
               float* __restrict__ out,          // [T] ++ h_n[40] ++ c_n[40]
               int T) {
    __shared__ float h0s[HID], h1s[HID], g0s[G4], g1s[G4];
    const int tid = threadIdx.x;

    float w0[HID];                 // layer0 recurrent row (tid < 80)
    float wi1[HID], wh1[HID];      // layer1 rows (tid in [96,176))
    float bias1 = 0.f;
    if (tid < G4) {
        #pragma unroll
        for (int k = 0; k < HID; ++k) w0[k] = W_hh0[tid * HID + k];
    }
    if (tid >= 96 && tid < 96 + G4) {
        int g = tid - 96;
        #pragma unroll
        for (int k = 0; k < HID; ++k) {
            wi1[k] = W_ih1[g * HID + k];
            wh1[k] = W_hh1[g * HID + k];
        }
        bias1 = b_ih1[g] + b_hh1[g];
    }
    float wout = (tid >= 32 && tid < 32 + HID) ? W_out[tid - 32] : 0.f;
    float bo = b_out[0];

    float c0r = 0.f, c1r = 0.f;
    if (tid < HID) { h0s[tid] = 0.f; h1s[tid] = 0.f; }
    __syncthreads();

    float pre_cur = (tid < G4 && T > 0) ? pre0[tid] : 0.f;

    for (int n = 0; n <= T; ++n) {
        // prefetch next pre0 row (one full step of latency to hide)
        float pre_nxt = 0.f;
        if (tid < G4 && (n + 1) < T) pre_nxt = pre0[(size_t)(n + 1) * G4 + tid];

        // S1a: layer0 gates for step n      (waves 0-2)
        if (tid < G4 && n < T) {
            float s = pre_cur;
            #pragma unroll
            for (int k = 0; k < HID; ++k) s = fmaf(w0[k], h0s[k], s);
            g0s[tid] = s;
        }
        // S1b: layer1 gates for step n-1    (waves 3-5, runs concurrently)
        if (tid >= 96 && tid < 96 + G4 && n > 0) {
            float s = bias1;
            #pragma unroll
            for (int k = 0; k < HID; ++k) s = fmaf(wi1[k], h0s[k], s);
            #pragma unroll
            for (int k = 0; k < HID; ++k) s = fmaf(wh1[k], h1s[k], s);
            g1s[tid - 96] = s;
        }
        __syncthreads();

        // S2a: layer0 cell update (wave 0 lanes 0-19)
        if (tid < HID && n < T) {
            float ig = sig_(g0s[tid]);
            float fg = sig_(g0s[tid + HID]);
            float gg = th_ (g0s[tid + 2 * HID]);
            float og = sig_(g0s[tid + 3 * HID]);
            c0r = fmaf(fg, c0r, ig * gg);
            h0s[tid] = og * th_(c0r);
        }
        // S2b: layer1 cell update for step n-1 + output reduction (wave 1)
        if (tid >= 32 && tid < 64 && n > 0) {
            int l = tid - 32;
            float p = 0.f;
            if (l < HID) {
                float ig = sig_(g1s[l]);
                float fg = sig_(g1s[l + HID]);
                float gg = th_ (g1s[l + 2 * HID]);
                float og = sig_(g1s[l + 3 * HID]);
                c1r = fmaf(fg, c1r, ig * gg);
                float h1v = og * th_(c1r);
                h1s[l] = h1v;
                p = h1v * wout;
            }
            #pragma unroll
            for (int m = 16; m >= 1; m >>= 1) p += __shfl_xor(p, m, 32);
            if (l == 0) out[n - 1] = p + bo;
        }
        __syncthreads();
        pre_cur = pre_nxt;
    }

    // final states: layout  outputs[T] ++ h_n[2][20] ++ c_n[2][20]
    if (tid < HID)                 { out[T + tid] = h0s[tid];          out[T + 2 * HID + tid] = c0r; }
    if (tid >= 32 && tid < 32 + HID) { int l = tid - 32;
                                     out[T + HID + l] = h1s[l];        out[T + 3 * HID + l]   = c1r; }
}

// ---------------------------------------------------------------------------
extern "C" void kernel_launch(void* const* d_in, const int* in_sizes, int n_in,
                              void* d_out, int out_size, void* d_ws, size_t ws_size,
                              hipStream_t stream) {
    const float* in_states = (const float*)d_in[0];
    const float* W_inp = (const float*)d_in[1];
    const float* b_inp = (const float*)d_in[2];
    const float* W_ih0 = (const float*)d_in[3];
    const float* W_hh0 = (const float*)d_in[4];
    const float* b_ih0 = (const float*)d_in[5];
    const float* b_hh0 = (const float*)d_in[6];
    const float* W_ih1 = (const float*)d_in[7];
    const float* W_hh1 = (const float*)d_in[8];
    const float* b_ih1 = (const float*)d_in[9];
    const float* b_hh1 = (const float*)d_in[10];
    const float* W_out = (const float*)d_in[11];
    const float* b_out = (const float*)d_in[12];
    float* out = (float*)d_out;

    int T = in_sizes[0] / T_IN;

    float* Wf   = (float*)d_ws;                     // 80*40 floats
    float* bf   = Wf + G4 * T_IN;                   // 80 floats
    float* pre0 = (float*)((char*)d_ws + 16384);    // T*80 floats (128 MB @ T=400k)

    fuse_weights<<<1, 128, 0, stream>>>(W_inp, b_inp, W_ih0, b_ih0, b_hh0, Wf, bf);

    int mtiles = (T + 15) / 16;
    int numTiles = mtiles * 5;                      // 5 N-tiles of 16 over 80 gates
    int blocks = (numTiles + 7) / 8;                // 8 waves per 256-thread block
    pre0_gemm<<<blocks, 256, 0, stream>>>(in_states, Wf, bf, pre0, T, numTiles);

    lstm_scan<<<1, 192, 0, stream>>>(pre0, W_hh0, W_ih1, W_hh1,
                                     b_ih1, b_hh1, W_out, b_out, out, T);
}